// ExampleModelSISO_19490561589730
// MI455X (gfx1250) — compile-verified
//
#include <hip/hip_runtime.h>

// out[b,t,d] = x[b,t,d] * exclusive_cumsum_t(x)[b,t,d]
// B=8, T=4096, D=1024, fp32. Memory-bound (256 MiB floor @ 23.3 TB/s ~= 11.5us).
// 3-pass chunked scan; pass C does the 16x16 intra-tile exclusive scan with
// V_WMMA_F32_16X16X4_F32 (Y = L_strict * X as 4 chained K=4 WMMAs).

typedef __attribute__((ext_vector_type(2))) float v2f;
typedef __attribute__((ext_vector_type(8))) float v8f;

constexpr int kB = 8;
constexpr int kT = 4096;
constexpr int kD = 1024;
constexpr int CHUNK = 16;                  // timesteps per tile / chunk
constexpr int NC = kT / CHUNK;             // 256 chunks per chain
constexpr int CHAINS = kB * kD;            // 8192 independent scan chains

// ---------------------------------------------------------------------------
// Pass A: S[c*CHAINS + chain] = sum of x over the 16 timesteps of chunk c.
// Consecutive threads -> consecutive d -> fully coalesced 128B/wave loads.
// 2M threads = 65536 waves of memory-level parallelism.
// ---------------------------------------------------------------------------
__global__ void chunk_sums(const float* __restrict__ x, float* __restrict__ S) {
    int i = blockIdx.x * blockDim.x + threadIdx.x;   // i = c*CHAINS + chain
    int chain = i % CHAINS;                          // chain = b*kD + d
    int c     = i / CHAINS;
    int b = chain / kD;
    int d = chain % kD;
    const float* p = x + ((size_t)(b * kT + c * CHUNK) * kD + d);
    float s = 0.0f;
#pragma unroll
    for (int t = 0; t < CHUNK; ++t) s += p[(size_t)t * kD];
    S[i] = s;
}

// ---------------------------------------------------------------------------
// Pass B: in-place exclusive scan of the NC chunk sums of each chain.
// One thread per chain; S layout [c][chain] keeps every access coalesced.
// Touches only 2 * 8 MiB -> negligible.
// ---------------------------------------------------------------------------
__global__ void scan_chunks(float* __restrict__ S) {
    int chain = blockIdx.x * blockDim.x + threadIdx.x;
    float run = 0.0f;
    for (int c = 0; c < NC; ++c) {
        size_t idx = (size_t)c * CHAINS + chain;
        float v = S[idx];
        S[idx] = run;
        run += v;
    }
}

// ---------------------------------------------------------------------------
// Pass C: one wave32 per 16(time) x 16(d) tile.
//   acc = carry_row (broadcast over M)                        [C operand]
//   acc = sum_j A_j(16x4) * B_j(4x16) + acc  via 4x WMMA      [K=16 total]
//     A_j[m][k] = 1 if (4j+k) < m else 0    (strict lower-triangular ones)
//     B_j       = rows 4j..4j+3 of X
//   out(M,N) = X(M,N) * acc(M,N)            (NT stores; output not re-read)
// X re-read here hits the 192MB L2 (whole input fits), not HBM.
// ISA 7.12.2 wave32 layouts:
//   A (16x4 f32): M = lane&15,  K = reg + 2*(lane>>4)
//   B (4x16 f32): N = lane&15,  K = reg + 2*(lane>>4)
//   C/D (16x16):  N = lane&15,  M = reg + 8*(lane>>4)
// ---------------------------------------------------------------------------
__global__ void scan_apply(const float* __restrict__ x,
                           const float* __restrict__ S,
                           float* __restrict__ out) {
    int wave = (blockIdx.x * blockDim.x + threadIdx.x) >> 5;
    int lane = threadIdx.x & 31;
    int n    = lane & 15;       // N (d within tile) for B/C/D; M for A
    int half = lane >> 4;

    int dblk = wave % (kD / 16);
    int tmp  = wave / (kD / 16);
    int c    = tmp % NC;
    int b    = tmp / NC;
    int d0   = dblk * 16;
    int t0   = c * CHUNK;

    // Per-column carry from pass B (lanes 16..31 re-read the same 16 floats;
    // these hit L0).
    float carry = S[(size_t)c * CHAINS + (size_t)b * kD + d0 + n];
    v8f acc;
#pragma unroll
    for (int i = 0; i < 8; ++i) acc[i] = carry;

    size_t base = (size_t)(b * kT + t0) * kD + d0 + n;   // element (t0, d0+n)
    const float* xb = x + base;

#pragma unroll
    for (int j = 0; j < 4; ++j) {
        v2f a, bm;
#pragma unroll
        for (int r = 0; r < 2; ++r) {
            int k = 4 * j + 2 * half + r;                // global K index
            a[r]  = (k < n) ? 1.0f : 0.0f;               // L_strict[m=n][k]
            bm[r] = xb[(size_t)k * kD];                  // X[k][n]
        }
        // (neg_a, A, neg_b, B, c_mod, C, reuse_a, reuse_b)
        acc = __builtin_amdgcn_wmma_f32_16x16x4_f32(
            false, a, false, bm, (short)0, acc, false, false);
    }

    // out(M,N) = x(M,N) * acc(M,N); M = i + 8*half. x re-read from L2/L0.
#pragma unroll
    for (int i = 0; i < 8; ++i) {
        size_t off = (size_t)(i + 8 * half) * kD;
        float xv = xb[off];
        __builtin_nontemporal_store(xv * acc[i], out + base + off);
    }
}

// ---------------------------------------------------------------------------
extern "C" void kernel_launch(void* const* d_in, const int* in_sizes, int n_in,
                              void* d_out, int out_size, void* d_ws, size_t ws_size,
                              hipStream_t stream) {
    (void)in_sizes; (void)n_in; (void)out_size; (void)ws_size;
    const float* x = (const float*)d_in[0];
    float* out = (float*)d_out;
    float* S   = (float*)d_ws;   // NC * CHAINS floats = 8 MiB scratch

    // Pass A: chunk sums. NC*CHAINS = 2,097,152 threads.
    chunk_sums<<<(NC * CHAINS) / 256, 256, 0, stream>>>(x, S);
    // Pass B: exclusive scan of chunk sums per chain.
    scan_chunks<<<CHAINS / 256, 256, 0, stream>>>(S);
    // Pass C: one wave per 16x16 tile. kB*NC*(kD/16) waves * 32 lanes.
    int wavesC = kB * NC * (kD / 16);                 // 131072 waves
    scan_apply<<<(wavesC * 32) / 256, 256, 0, stream>>>(x, S, out);
}